// TreeEncoder_16458314678344
// MI455X (gfx1250) — compile-verified
//
#include <hip/hip_runtime.h>
#include <hip/hip_bf16.h>

typedef float v2f __attribute__((ext_vector_type(2)));
typedef float v8f __attribute__((ext_vector_type(8)));

#define CIN  16
#define HID  32
#define COUT 64
#define K9   9

// ---------------------------------------------------------------------------
// Stage 1: h = relu(gather(feat, neigh_child) @ W1 + b1)     [N,32]
// One wave per 32-node tile (two 16-row A fragments -> each B fragment feeds
// 2 WMMAs). K=144 -> 36 chunks of 4 via v_wmma_f32_16x16x4_f32.
// W1 staged in LDS k-paired: sW[(k/2)*32 + n][2] -> B fragment = 1 ds_load_b64.
// ---------------------------------------------------------------------------
__global__ __launch_bounds__(256) void conv1_kernel(
    const float* __restrict__ feat,    // [N,16]
    const int*   __restrict__ neigh,   // [N,9]
    const float* __restrict__ W1,      // [144,32]
    const float* __restrict__ b1,      // [32]
    const float* __restrict__ zpad,    // >=16 zero floats
    float*       __restrict__ h,       // [N,32]
    int N)
{
    __shared__ float sW[K9 * CIN * HID];   // 18 KB, k-paired layout
    const int tid = threadIdx.x;
    for (int i = tid; i < K9 * CIN * HID; i += 256) {
        const int k = i >> 5, n = i & 31;
        sW[((k >> 1) * HID + n) * 2 + (k & 1)] = W1[i];
    }
    __syncthreads();

    const int wave = tid >> 5;
    const int lane = tid & 31;
    const int hb   = lane >> 4;          // lane half (0/1)
    const int m    = lane & 15;
    const int tile = blockIdx.x * 8 + wave;   // 32-node tile
    const int base = tile * 32;
    const int rowA = base + m;           // first 16-row A tile
    const int rowB = base + 16 + m;      // second 16-row A tile

    // preload neighbor indices for both rows; resolve -1 -> zero-pad pointer
    const float* srowA[K9];
    const float* srowB[K9];
#pragma unroll
    for (int j = 0; j < K9; ++j) {
        const int ia = neigh[rowA * K9 + j];
        const int ib = neigh[rowB * K9 + j];
        srowA[j] = (ia < 0) ? zpad : (feat + (size_t)ia * CIN);
        srowB[j] = (ib < 0) ? zpad : (feat + (size_t)ib * CIN);
    }

    v8f acc00 = {}, acc01 = {};    // rows base..base+15, col tiles 0/1
    v8f acc10 = {}, acc11 = {};    // rows base+16..base+31

#pragma unroll
    for (int j = 0; j < K9; ++j) {
#pragma unroll
        for (int kk = 0; kk < CIN / 4; ++kk) {     // 4 k-chunks per neighbor
            const int cbase = kk * 4 + 2 * hb;     // channel pair for this lane
            const v2f a0 = *(const v2f*)(srowA[j] + cbase);
            const v2f a1 = *(const v2f*)(srowB[j] + cbase);

            const int k2 = (j * CIN / 2) + kk * 2 + hb;   // paired-k index
            const v2f bf0 = *(const v2f*)(sW + (k2 * HID + m) * 2);
            const v2f bf1 = *(const v2f*)(sW + (k2 * HID + 16 + m) * 2);

            acc00 = __builtin_amdgcn_wmma_f32_16x16x4_f32(
                false, a0, false, bf0, (short)0, acc00, false, false);
            acc01 = __builtin_amdgcn_wmma_f32_16x16x4_f32(
                false, a0, false, bf1, (short)0, acc01, false, false);
            acc10 = __builtin_amdgcn_wmma_f32_16x16x4_f32(
                false, a1, false, bf0, (short)0, acc10, false, false);
            acc11 = __builtin_amdgcn_wmma_f32_16x16x4_f32(
                false, a1, false, bf1, (short)0, acc11, false, false);
        }
    }

    const float bias0 = b1[m];
    const float bias1 = b1[16 + m];
#pragma unroll
    for (int i = 0; i < 8; ++i) {
        const int rA = base + i + 8 * hb;        // C/D layout: vgpr i -> M=i+8*half
        const int rB = base + 16 + i + 8 * hb;
        float v;
        v = acc00[i] + bias0; h[(size_t)rA * HID + m]      = v > 0.f ? v : 0.f;
        v = acc01[i] + bias1; h[(size_t)rA * HID + 16 + m] = v > 0.f ? v : 0.f;
        v = acc10[i] + bias0; h[(size_t)rB * HID + m]      = v > 0.f ? v : 0.f;
        v = acc11[i] + bias1; h[(size_t)rB * HID + 16 + m] = v > 0.f ? v : 0.f;
    }
}

// ---------------------------------------------------------------------------
// Stage 2: p = masked mean over up-to-4 children.   [NP,32]
// 8 threads per parent, each handles a float4 of channels.
// ---------------------------------------------------------------------------
__global__ __launch_bounds__(256) void pool_kernel(
    const float* __restrict__ h,          // [N,32]
    const int*   __restrict__ children,   // [NP,4]
    const float* __restrict__ zpad,       // >=32 zero floats
    float*       __restrict__ p,          // [NP,32]
    int NP)
{
    const int t = blockIdx.x * 256 + threadIdx.x;
    if (t >= NP * 8) return;
    const int parent = t >> 3;
    const int c4     = (t & 7) * 4;

    const int4 ci = *(const int4*)(children + (size_t)parent * 4);
    const int  cid[4] = {ci.x, ci.y, ci.z, ci.w};

    float sx = 0.f, sy = 0.f, sz = 0.f, sw = 0.f, cnt = 0.f;
#pragma unroll
    for (int j = 0; j < 4; ++j) {
        const int id = cid[j];
        const float* src = (id < 0) ? zpad : (h + (size_t)id * HID + c4);
        const float4 v = *(const float4*)src;
        sx += v.x; sy += v.y; sz += v.z; sw += v.w;
        cnt += (id < 0) ? 0.f : 1.f;
    }
    cnt = cnt < 1.0f ? 1.0f : cnt;
    const float inv = 1.0f / cnt;
    float4 o; o.x = sx * inv; o.y = sy * inv; o.z = sz * inv; o.w = sw * inv;
    *(float4*)(p + (size_t)parent * HID + c4) = o;
}

// ---------------------------------------------------------------------------
// Stage 3: out = gather(p, neigh_parent) @ W2 + b2     [NP,64]
// 32-row tile per wave; K=288 -> 72 chunks, COUT=64 -> 4 column tiles.
// W2 (72 KB) staged via DYNAMIC LDS (320 KB/WGP on CDNA5), k-paired layout.
// ---------------------------------------------------------------------------
__global__ __launch_bounds__(256) void conv2_kernel(
    const float* __restrict__ p,       // [NP,32]
    const int*   __restrict__ neigh,   // [NP,9]
    const float* __restrict__ W2,      // [288,64]
    const float* __restrict__ b2,      // [64]
    const float* __restrict__ zpad,    // >=32 zero floats
    float*       __restrict__ out,     // [NP,64]
    int NP)
{
    extern __shared__ float sW2[];     // 288*64 floats = 72 KB (dynamic)
    const int tid = threadIdx.x;
    for (int i = tid; i < K9 * HID * COUT; i += 256) {
        const int k = i >> 6, n = i & 63;
        sW2[((k >> 1) * COUT + n) * 2 + (k & 1)] = W2[i];
    }
    __syncthreads();

    const int wave = tid >> 5;
    const int lane = tid & 31;
    const int hb   = lane >> 4;
    const int m    = lane & 15;
    const int tile = blockIdx.x * 8 + wave;
    const int base = tile * 32;
    const int rowA = base + m;
    const int rowB = base + 16 + m;

    const float* srowA[K9];
    const float* srowB[K9];
#pragma unroll
    for (int j = 0; j < K9; ++j) {
        const int ia = neigh[rowA * K9 + j];
        const int ib = neigh[rowB * K9 + j];
        srowA[j] = (ia < 0) ? zpad : (p + (size_t)ia * HID);
        srowB[j] = (ib < 0) ? zpad : (p + (size_t)ib * HID);
    }

    v8f accA[4] = {{}, {}, {}, {}};   // rows base..base+15
    v8f accB[4] = {{}, {}, {}, {}};   // rows base+16..base+31

#pragma unroll
    for (int j = 0; j < K9; ++j) {
        for (int kk = 0; kk < HID / 4; ++kk) {     // 8 k-chunks per neighbor
            const int cbase = kk * 4 + 2 * hb;
            const v2f a0 = *(const v2f*)(srowA[j] + cbase);
            const v2f a1 = *(const v2f*)(srowB[j] + cbase);

            const int k2 = (j * HID / 2) + kk * 2 + hb;
#pragma unroll
            for (int nt = 0; nt < 4; ++nt) {
                const v2f bf = *(const v2f*)(sW2 + (k2 * COUT + nt * 16 + m) * 2);
                accA[nt] = __builtin_amdgcn_wmma_f32_16x16x4_f32(
                    false, a0, false, bf, (short)0, accA[nt], false, false);
                accB[nt] = __builtin_amdgcn_wmma_f32_16x16x4_f32(
                    false, a1, false, bf, (short)0, accB[nt], false, false);
            }
        }
    }

#pragma unroll
    for (int nt = 0; nt < 4; ++nt) {
        const float bias = b2[nt * 16 + m];
#pragma unroll
        for (int i = 0; i < 8; ++i) {
            const int rA = base + i + 8 * hb;
            const int rB = base + 16 + i + 8 * hb;
            out[(size_t)rA * COUT + nt * 16 + m] = accA[nt][i] + bias;
            out[(size_t)rB * COUT + nt * 16 + m] = accB[nt][i] + bias;
        }
    }
}

// ---------------------------------------------------------------------------
extern "C" void kernel_launch(void* const* d_in, const int* in_sizes, int n_in,
                              void* d_out, int out_size, void* d_ws, size_t ws_size,
                              hipStream_t stream)
{
    const float* feat         = (const float*)d_in[0];
    const float* W1           = (const float*)d_in[1];
    const float* b1           = (const float*)d_in[2];
    const float* W2           = (const float*)d_in[3];
    const float* b2           = (const float*)d_in[4];
    const int*   neigh_child  = (const int*)d_in[5];
    const int*   neigh_parent = (const int*)d_in[6];
    const int*   children_idx = (const int*)d_in[7];

    const int N  = in_sizes[0] / CIN;      // 524288
    const int NP = in_sizes[7] / 4;        // 131072

    // workspace layout: [zpad 32 floats][h N*32][p NP*32]
    float* zpad = (float*)d_ws;
    float* h    = zpad + 32;               // [N,32]   64 MB
    float* pp   = h + (size_t)N * HID;     // [NP,32]  16 MB
    float* out  = (float*)d_out;           // [NP,64]

    hipMemsetAsync(zpad, 0, 32 * sizeof(float), stream);   // capture-safe

    // 8 waves/block, one 32-node tile per wave -> 256 nodes per block
    conv1_kernel<<<N / 256, 256, 0, stream>>>(feat, neigh_child, W1, b1, zpad, h, N);
    pool_kernel<<<(NP * 8 + 255) / 256, 256, 0, stream>>>(h, children_idx, zpad, pp, NP);
    conv2_kernel<<<NP / 256, 256, (size_t)(K9 * HID * COUT * sizeof(float)), stream>>>(
        pp, neigh_parent, W2, b2, zpad, out, NP);
}